// BFA_67997922230423
// MI455X (gfx1250) — compile-verified
//
#include <hip/hip_runtime.h>
#include <hip/hip_bf16.h>

// Canny(gray) + concat + 1x1 conv (K=4) + bias + ReLU for MI455X (gfx1250).
// The 1x1 conv is mapped onto V_WMMA_F32_16X16X4_F32 (K=4 == C+1 exactly).
// Output (256 MB, write-once) is stored with non-temporal hints to keep the
// 192 MB L2 free for the re-used input planes / Canny intermediates.

typedef __attribute__((ext_vector_type(2))) float v2f;
typedef __attribute__((ext_vector_type(8))) float v8f;

#define BB 4
#define HH 512
#define WW 512
#define HW (HH * WW)
#define NPIX (BB * HW)

// ---------------- Stage 1: grayscale (emulate uint8 cast) -------------------
__global__ void BFA_gray_k(const float* __restrict__ x, float* __restrict__ gray) {
    int i = blockIdx.x * 256 + threadIdx.x;          // over B*HW
    int b = i / HW, hw = i % HW;
    const float* xb = x + (long)b * 3 * HW;
    float g = 0.2989f * xb[hw] + 0.587f * xb[HW + hw] + 0.114f * xb[2 * HW + hw];
    g = fminf(fmaxf(g, 0.0f), 255.0f);
    gray[i] = truncf(g);
}

// ---------------- Stage 2: Sobel + |grad| + direction class -----------------
__global__ void BFA_sobel_k(const float* __restrict__ gray, float* __restrict__ mag,
                            unsigned char* __restrict__ dir) {
    int i = blockIdx.x * 256 + threadIdx.x;
    int b = i / HW, hw = i % HW;
    int y = hw / WW, xx = hw % WW;
    const float* g = gray + (long)b * HW;
    auto R = [&](int yy, int xc) {
        yy = yy < 0 ? -yy : (yy >= HH ? 2 * HH - 2 - yy : yy);   // REFLECT_101
        xc = xc < 0 ? -xc : (xc >= WW ? 2 * WW - 2 - xc : xc);
        return g[yy * WW + xc];
    };
    float a00 = R(y - 1, xx - 1), a01 = R(y - 1, xx), a02 = R(y - 1, xx + 1);
    float a10 = R(y,     xx - 1),                     a12 = R(y,     xx + 1);
    float a20 = R(y + 1, xx - 1), a21 = R(y + 1, xx), a22 = R(y + 1, xx + 1);
    float gx = (a02 - a00) + 2.0f * (a12 - a10) + (a22 - a20);
    float gy = (a20 - a00) + 2.0f * (a21 - a01) + (a22 - a02);
    mag[i] = fabsf(gx) + fabsf(gy);
    float ang = atan2f(gy, gx) * 57.29577951308232f;   // degrees
    ang = fmodf(ang, 180.0f);
    if (ang < 0.0f) ang += 180.0f;                     // jnp.mod semantics
    unsigned char d;
    if (ang < 22.5f || ang >= 157.5f) d = 0;
    else if (ang < 67.5f)             d = 1;
    else if (ang < 112.5f)            d = 2;
    else                              d = 3;
    dir[i] = d;
}

// ---------------- Stage 3: NMS + double threshold ---------------------------
__global__ void BFA_nms_k(const float* __restrict__ mag, const unsigned char* __restrict__ dir,
                          unsigned char* __restrict__ sb, unsigned char* __restrict__ wb) {
    int i = blockIdx.x * 256 + threadIdx.x;
    int b = i / HW, hw = i % HW;
    int y = hw / WW, xx = hw % WW;
    const float* m = mag + (long)b * HW;
    auto P = [&](int yy, int xc) {     // zero padding (matches jnp.pad default)
        return (yy < 0 || yy >= HH || xc < 0 || xc >= WW) ? 0.0f : m[yy * WW + xc];
    };
    float mv = m[hw];
    unsigned char d = dir[i];
    float n1, n2;
    if (d == 0)      { n1 = P(y, xx + 1);     n2 = P(y, xx - 1); }
    else if (d == 1) { n1 = P(y - 1, xx + 1); n2 = P(y + 1, xx - 1); }
    else if (d == 2) { n1 = P(y + 1, xx);     n2 = P(y - 1, xx); }
    else             { n1 = P(y - 1, xx - 1); n2 = P(y + 1, xx + 1); }
    float nms = (mv >= n1 && mv >= n2) ? mv : 0.0f;
    sb[i] = nms > 150.0f ? 1 : 0;    // strong (seed)
    wb[i] = nms > 50.0f ? 1 : 0;     // weak (growth medium; strong subset of weak)
}

// ---------------- Stage 4: hysteresis (tiled in-LDS fixed point) ------------
#define TS 64
__global__ void BFA_hyst_k(unsigned char* __restrict__ s, const unsigned char* __restrict__ wk) {
    __shared__ unsigned char st[(TS + 2) * (TS + 2)];
    __shared__ unsigned char wt[TS * TS];
    __shared__ int changed;
    const int tx = blockIdx.x * TS, ty = blockIdx.y * TS, b = blockIdx.z;
    const long base = (long)b * HW;
    const int tid = threadIdx.x;
    for (int i = tid; i < (TS + 2) * (TS + 2); i += 256) {
        int r = i / (TS + 2) - 1, c = i % (TS + 2) - 1;
        int gy = ty + r, gx = tx + c;
        unsigned char v = 0;
        if (gy >= 0 && gy < HH && gx >= 0 && gx < WW) v = s[base + gy * WW + gx];
        st[i] = v;
    }
    for (int i = tid; i < TS * TS; i += 256)
        wt[i] = wk[base + (ty + i / TS) * WW + tx + i % TS];
    if (tid == 0) changed = 0;
    __syncthreads();
    for (int it = 0; it < 1024; ++it) {
        // sweep: weak pixel with any 8-neighbor strong becomes strong (monotone)
        #pragma unroll
        for (int k = 0; k < (TS * TS) / 256; ++k) {
            int cell = tid + k * 256;
            int r = cell / TS, c = cell % TS;
            int li = (r + 1) * (TS + 2) + (c + 1);
            const int P = TS + 2;
            if (!st[li] && wt[cell]) {
                int nb = st[li - 1] | st[li + 1] | st[li - P] | st[li + P] |
                         st[li - P - 1] | st[li - P + 1] | st[li + P - 1] | st[li + P + 1];
                if (nb) { st[li] = 1; changed = 1; }
            }
        }
        __syncthreads();
        int any = changed;
        __syncthreads();
        if (tid == 0) changed = 0;
        __syncthreads();
        if (!any) break;
    }
    for (int i = tid; i < TS * TS; i += 256)
        s[base + (ty + i / TS) * WW + tx + i % TS] = st[(i / TS + 1) * (TS + 2) + i % TS + 1];
}

// ---------------- Stage 5: 1x1 conv via V_WMMA_F32_16X16X4_F32 --------------
// D(16 outch x 16 pix) = A(Wc 16x4) * B(cat 4x16) + C(bias). OUT=64 -> 4 WMMAs.
__global__ void BFA_conv_wmma_k(const float* __restrict__ x, const float* __restrict__ Wc,
                                const float* __restrict__ bias, const unsigned char* __restrict__ s,
                                float* __restrict__ out) {
    const int wave = threadIdx.x >> 5;
    const int lane = threadIdx.x & 31;
    const long g0 = (long)blockIdx.x * 128 + wave * 16;   // 16-pixel group over B*HW
    const int b = (int)(g0 / HW);
    const int hw = (int)(g0 % HW);
    const int half = lane >> 4;           // 0: lanes 0-15, 1: lanes 16-31
    const int pix = hw + (lane & 15);

    // B-matrix 4x16 f32: lane%16 = N (pixel); lanes 0-15 -> K rows 0,1 in v0,v1;
    // lanes 16-31 -> K rows 2,3. Channel 3 = Canny edge (s ? 255 : 0).
    v2f bm;
    if (half == 0) {
        bm.x = x[(long)(b * 3 + 0) * HW + pix];
        bm.y = x[(long)(b * 3 + 1) * HW + pix];
    } else {
        bm.x = x[(long)(b * 3 + 2) * HW + pix];
        bm.y = s[(long)b * HW + pix] ? 255.0f : 0.0f;
    }

    #pragma unroll
    for (int m = 0; m < 4; ++m) {
        // A-matrix 16x4 f32 (ISA-documented): lane%16 = M row; lane half selects K pair.
        int row = m * 16 + (lane & 15);
        int k0 = half * 2;
        v2f am;
        am.x = Wc[row * 4 + k0];
        am.y = Wc[row * 4 + k0 + 1];
        // C = bias, laid out per documented f32 C/D tiling (VGPR r -> M = r + 8*half).
        v8f c;
        #pragma unroll
        for (int r = 0; r < 8; ++r) c[r] = bias[m * 16 + r + half * 8];
        v8f d = __builtin_amdgcn_wmma_f32_16x16x4_f32(
            /*neg_a=*/false, am, /*neg_b=*/false, bm,
            /*c_mod=*/(short)0, c, /*reuse_a=*/false, /*reuse_b=*/false);
        #pragma unroll
        for (int r = 0; r < 8; ++r) {
            int o = m * 16 + r + half * 8;
            float v = d[r];
            v = v > 0.0f ? v : 0.0f;                      // ReLU
            // Write-once 256 MB stream: non-temporal store keeps L2 for reads.
            __builtin_nontemporal_store(v, &out[(long)(b * 64 + o) * HW + pix]);
        }
    }
}

// ---------------------------------------------------------------------------
extern "C" void kernel_launch(void* const* d_in, const int* in_sizes, int n_in,
                              void* d_out, int out_size, void* d_ws, size_t ws_size,
                              hipStream_t stream) {
    const float* x    = (const float*)d_in[0];   // [4,3,512,512]
    const float* Wc   = (const float*)d_in[1];   // [64,4]
    const float* bias = (const float*)d_in[2];   // [64]
    float* out = (float*)d_out;                  // [4,64,512,512]

    char* ws = (char*)d_ws;
    float* gray        = (float*)ws;                                   // 4 MB
    float* mag         = (float*)(ws + (size_t)NPIX * 4);              // 4 MB
    unsigned char* dir = (unsigned char*)(ws + (size_t)NPIX * 8);      // 1 MB
    unsigned char* sb  = dir + NPIX;                                   // 1 MB
    unsigned char* wb  = sb + NPIX;                                    // 1 MB

    BFA_gray_k <<<NPIX / 256, 256, 0, stream>>>(x, gray);
    BFA_sobel_k<<<NPIX / 256, 256, 0, stream>>>(gray, mag, dir);
    BFA_nms_k  <<<NPIX / 256, 256, 0, stream>>>(mag, dir, sb, wb);
    // Each pass converges a 64x64 tile in LDS; 48 grid-wide passes carry the
    // frontier across tile boundaries (monotone -> unique fixed point).
    for (int p = 0; p < 48; ++p)
        BFA_hyst_k<<<dim3(WW / TS, HH / TS, BB), 256, 0, stream>>>(sb, wb);
    BFA_conv_wmma_k<<<NPIX / 128, 256, 0, stream>>>(x, Wc, bias, sb, out);
}